// PARgg_81561428951544
// MI455X (gfx1250) — compile-verified
//
#include <hip/hip_runtime.h>
#include <math.h>

#define HH 224
#define WW 224
#define HW 50176
#define H2 67
#define W2 67
#define HW2 4489
#define CIMG 3
#define CMSK 21
#define KN 48
#define EPSV 1e-8f
#define W1V 0.3f
#define NITER 10

// 8-neighborhood displacements per dilation: (ki,kj) in
// {(0,0),(0,1),(0,2),(1,0),(1,2),(2,0),(2,1),(2,2)}, disp = d*(k-1)
#define DY8(d) -(d), -(d), -(d), 0, 0, (d), (d), (d)
#define DX8(d) -(d), 0, (d), -(d), (d), -(d), 0, (d)
__device__ constexpr int k_dy[KN] = {DY8(1), DY8(2), DY8(4), DY8(8), DY8(12), DY8(24)};
__device__ constexpr int k_dx[KN] = {DX8(1), DX8(2), DX8(4), DX8(8), DX8(12), DX8(24)};

__device__ __forceinline__ int clampi(int v, int hi) {
    return v < 0 ? 0 : (v > hi ? hi : v);
}

typedef __attribute__((ext_vector_type(2))) float v2f;
typedef __attribute__((ext_vector_type(8))) float v8f;

// ---------------------------------------------------------------------------
// Generic f32 WMMA GEMM: C[b] = A[b](MxK) * B[b](KxN), one wave per 16x16 tile.
// stride==0 means the operand is shared across the batch dimension.
// A 16x4 f32 fragment: lane L holds M=L&15, K = r + 2*(L>>4) in VGPR r.
// B 4x16 f32 fragment: lane L holds N=L&15, K = r + 2*(L>>4) in VGPR r.
// C/D 16x16: VGPR v, lane L -> (M = v + 8*(L>>4), N = L&15).
// ---------------------------------------------------------------------------
__global__ __launch_bounds__(32) void gemm_wmma_f32(
    const float* __restrict__ A, long sA,
    const float* __restrict__ B, long sB,
    float* __restrict__ C, long sC,
    int M, int N, int Kd, int lda, int ldb, int ldc) {
    const int lane = threadIdx.x;
    const int half = lane >> 4;
    const int l16  = lane & 15;
    const int m = blockIdx.x * 16 + l16;
    const int n = blockIdx.y * 16 + l16;
    const int b = blockIdx.z;
    const float* Ab = A + (size_t)b * (size_t)sA;
    const float* Bb = B + (size_t)b * (size_t)sB;
    float* Cb = C + (size_t)b * (size_t)sC;

    const bool mok = (m < M);
    const bool nok = (n < N);
    const int  mi  = mok ? m : (M - 1);
    const int  ni  = nok ? n : (N - 1);

    v8f acc = {};
    for (int kk = 0; kk < Kd; kk += 4) {
        const int ka0 = kk + 2 * half;
        const int ka1 = ka0 + 1;
        const int k0c = ka0 < Kd ? ka0 : (Kd - 1);
        const int k1c = ka1 < Kd ? ka1 : (Kd - 1);
        const float fa0 = (mok && ka0 < Kd) ? 1.f : 0.f;
        const float fa1 = (mok && ka1 < Kd) ? 1.f : 0.f;
        const float fb0 = (nok && ka0 < Kd) ? 1.f : 0.f;
        const float fb1 = (nok && ka1 < Kd) ? 1.f : 0.f;
        v2f av, bv;
        av.x = Ab[(size_t)mi * lda + k0c] * fa0;
        av.y = Ab[(size_t)mi * lda + k1c] * fa1;
        bv.x = Bb[(size_t)k0c * ldb + ni] * fb0;
        bv.y = Bb[(size_t)k1c * ldb + ni] * fb1;
        acc = __builtin_amdgcn_wmma_f32_16x16x4_f32(
            false, av, false, bv, (short)0, acc, false, false);
    }
#pragma unroll
    for (int v = 0; v < 8; ++v) {
        const int row = blockIdx.x * 16 + v + 8 * half;
        if (row < M && nok) Cb[(size_t)row * ldc + n] = acc[v];
    }
}

// ---------------------------------------------------------------------------
__global__ void zero_kernel(float* __restrict__ p, long n) {
    long i = (long)blockIdx.x * blockDim.x + threadIdx.x;
    if (i < n) p[i] = 0.f;
}

// Build bilinear (align_corners=True) resize matrix R (oh x ih) and RT (ih x oh).
__global__ void fill_resize(float* __restrict__ R, float* __restrict__ RT,
                            int oh, int ih) {
    int t = blockIdx.x * blockDim.x + threadIdx.x;
    if (t >= oh) return;
    float y = (float)t * (float)(ih - 1) / (float)(oh - 1);
    int   y0 = (int)floorf(y);
    int   y1 = min(y0 + 1, ih - 1);
    float wy = y - (float)y0;
    R[t * ih + y0] += 1.f - wy;
    R[t * ih + y1] += wy;
    RT[y0 * oh + t] += 1.f - wy;
    RT[y1 * oh + t] += wy;
}

// Softmax of the positional affinity vector (48 values) -> ws.
__global__ void pos_kernel(float* __restrict__ pos_out) {
    if (threadIdx.x != 0 || blockIdx.x != 0) return;
    const float s2 = 1.41421356237309515f;
    const int   dil[6] = {1, 2, 4, 8, 12, 24};
    float p[KN];
    for (int di = 0; di < 6; ++di)
        for (int jj = 0; jj < 8; ++jj) {
            float base = (jj == 0 || jj == 2 || jj == 5 || jj == 7) ? s2 : 1.f;
            p[di * 8 + jj] = base * (float)dil[di];
        }
    float s = 0.f, q = 0.f;
    for (int k = 0; k < KN; ++k) { s += p[k]; q += p[k] * p[k]; }
    float var = (q - s * s / (float)KN) / (float)(KN - 1);
    float sd  = sqrtf(fmaxf(var, 0.f));
    float a[KN], mx = -3.4e38f;
    for (int k = 0; k < KN; ++k) {
        float t = p[k] / (sd + EPSV) / W1V;
        a[k] = -t * t;
        mx = fmaxf(mx, a[k]);
    }
    float se = 0.f;
    for (int k = 0; k < KN; ++k) { a[k] = expf(a[k] - mx); se += a[k]; }
    for (int k = 0; k < KN; ++k) pos_out[k] = a[k] / se;
}

// ---------------------------------------------------------------------------
// Full-res affinity: per pixel, std over 48 neighbors per channel (ddof=1),
// aff = mean_c(-(|v-u|/(std+eps)/W1)^2), softmax over K, write s1 -> aff_total.
// ---------------------------------------------------------------------------
__global__ __launch_bounds__(256) void aff_full_kernel(
    const float* __restrict__ img, float* __restrict__ s1out) {
    int p = blockIdx.x * 256 + threadIdx.x;
    if (p >= HW) return;
    const int i = p / WW, j = p % WW;
    const float u0 = img[0 * HW + p], u1 = img[1 * HW + p], u2 = img[2 * HW + p];
    float s0 = 0.f, s1 = 0.f, s2 = 0.f, q0 = 0.f, q1 = 0.f, q2 = 0.f;
#pragma unroll
    for (int k = 0; k < KN; ++k) {
        const int q = clampi(i + k_dy[k], HH - 1) * WW + clampi(j + k_dx[k], WW - 1);
        const float v0 = img[0 * HW + q], v1 = img[1 * HW + q], v2 = img[2 * HW + q];
        s0 += v0; q0 += v0 * v0;
        s1 += v1; q1 += v1 * v1;
        s2 += v2; q2 += v2 * v2;
    }
    const float rk = 1.f / KN, rk1 = 1.f / (KN - 1);
    const float i0 = 1.f / ((sqrtf(fmaxf((q0 - s0 * s0 * rk) * rk1, 0.f)) + EPSV) * W1V);
    const float i1 = 1.f / ((sqrtf(fmaxf((q1 - s1 * s1 * rk) * rk1, 0.f)) + EPSV) * W1V);
    const float i2 = 1.f / ((sqrtf(fmaxf((q2 - s2 * s2 * rk) * rk1, 0.f)) + EPSV) * W1V);
    float a[KN], mx = -3.4e38f;
#pragma unroll
    for (int k = 0; k < KN; ++k) {
        const int q = clampi(i + k_dy[k], HH - 1) * WW + clampi(j + k_dx[k], WW - 1);
        const float d0 = fabsf(img[0 * HW + q] - u0) * i0;
        const float d1 = fabsf(img[1 * HW + q] - u1) * i1;
        const float d2 = fabsf(img[2 * HW + q] - u2) * i2;
        a[k] = -(d0 * d0 + d1 * d1 + d2 * d2) * (1.f / 3.f);
        mx = fmaxf(mx, a[k]);
    }
    float se = 0.f;
#pragma unroll
    for (int k = 0; k < KN; ++k) { a[k] = expf(a[k] - mx); se += a[k]; }
    const float rs = 1.f / se;
#pragma unroll
    for (int k = 0; k < KN; ++k) s1out[(size_t)k * HW + p] = a[k] * rs;
}

// Low-res affinity (no /W1, NO softmax here — softmax happens after upsample).
__global__ __launch_bounds__(256) void aff_low_kernel(
    const float* __restrict__ img, float* __restrict__ affout) {
    int p = blockIdx.x * 256 + threadIdx.x;
    if (p >= HW2) return;
    const int i = p / W2, j = p % W2;
    const float u0 = img[0 * HW2 + p], u1 = img[1 * HW2 + p], u2 = img[2 * HW2 + p];
    float s0 = 0.f, s1 = 0.f, s2 = 0.f, q0 = 0.f, q1 = 0.f, q2 = 0.f;
#pragma unroll
    for (int k = 0; k < KN; ++k) {
        const int q = clampi(i + k_dy[k], H2 - 1) * W2 + clampi(j + k_dx[k], W2 - 1);
        const float v0 = img[0 * HW2 + q], v1 = img[1 * HW2 + q], v2 = img[2 * HW2 + q];
        s0 += v0; q0 += v0 * v0;
        s1 += v1; q1 += v1 * v1;
        s2 += v2; q2 += v2 * v2;
    }
    const float rk = 1.f / KN, rk1 = 1.f / (KN - 1);
    const float i0 = 1.f / (sqrtf(fmaxf((q0 - s0 * s0 * rk) * rk1, 0.f)) + EPSV);
    const float i1 = 1.f / (sqrtf(fmaxf((q1 - s1 * s1 * rk) * rk1, 0.f)) + EPSV);
    const float i2 = 1.f / (sqrtf(fmaxf((q2 - s2 * s2 * rk) * rk1, 0.f)) + EPSV);
#pragma unroll
    for (int k = 0; k < KN; ++k) {
        const int q = clampi(i + k_dy[k], H2 - 1) * W2 + clampi(j + k_dx[k], W2 - 1);
        const float d0 = fabsf(img[0 * HW2 + q] - u0) * i0;
        const float d1 = fabsf(img[1 * HW2 + q] - u1) * i1;
        const float d2 = fabsf(img[2 * HW2 + q] - u2) * i2;
        affout[(size_t)k * HW2 + p] = -(d0 * d0 + d1 * d1 + d2 * d2) * (1.f / 3.f);
    }
}

// combine: aff_total[k,p] += softmax_k( bilinear_up(aff2)[k,p] ) + pos_soft[k]
__global__ __launch_bounds__(256) void combine_kernel(
    const float* __restrict__ aff2, const float* __restrict__ pos,
    float* __restrict__ afft) {
    int p = blockIdx.x * 256 + threadIdx.x;
    if (p >= HW) return;
    const int i = p / WW, j = p % WW;
    const float fy = (float)i * (float)(H2 - 1) / (float)(HH - 1);
    const float fx = (float)j * (float)(W2 - 1) / (float)(WW - 1);
    const int y0 = (int)floorf(fy), x0 = (int)floorf(fx);
    const int y1 = min(y0 + 1, H2 - 1), x1 = min(x0 + 1, W2 - 1);
    const float wy = fy - (float)y0, wx = fx - (float)x0;
    const float w00 = (1.f - wy) * (1.f - wx), w10 = wy * (1.f - wx);
    const float w01 = (1.f - wy) * wx,         w11 = wy * wx;
    const int o00 = y0 * W2 + x0, o10 = y1 * W2 + x0;
    const int o01 = y0 * W2 + x1, o11 = y1 * W2 + x1;
    float a[KN], mx = -3.4e38f;
#pragma unroll
    for (int k = 0; k < KN; ++k) {
        const float* b = aff2 + (size_t)k * HW2;
        a[k] = b[o00] * w00 + b[o10] * w10 + b[o01] * w01 + b[o11] * w11;
        mx = fmaxf(mx, a[k]);
    }
    float se = 0.f;
#pragma unroll
    for (int k = 0; k < KN; ++k) { a[k] = expf(a[k] - mx); se += a[k]; }
    const float rs = 1.f / se;
#pragma unroll
    for (int k = 0; k < KN; ++k)
        afft[(size_t)k * HW + p] += a[k] * rs + pos[k];
}

// ---------------------------------------------------------------------------
// Propagation: 48-tap weighted gather. 16x16 pixel tile per block; the 48
// per-pixel weights live in registers across the 21-channel loop; mask taps
// come from a 64x64 LDS halo tile (halo 24, edge-replicated via clamp).
// ---------------------------------------------------------------------------
__global__ __launch_bounds__(256) void prop_kernel(
    const float* __restrict__ src, const float* __restrict__ aff,
    float* __restrict__ dst) {
    __shared__ float sM[64 * 64];
    const int t  = threadIdx.x;
    const int tx = t & 15, ty = t >> 4;
    const int bx = blockIdx.x * 16, by = blockIdx.y * 16;
    const int gi = by + ty, gj = bx + tx;
    const int pix = gi * WW + gj;

    float wk[KN];
#pragma unroll
    for (int k = 0; k < KN; ++k) wk[k] = aff[(size_t)k * HW + pix];

    for (int c = 0; c < CMSK; ++c) {
        __syncthreads();
        const float* sc = src + (size_t)c * HW;
        for (int e = t; e < 64 * 64; e += 256) {
            const int ly = e >> 6, lx = e & 63;
            const int yy = clampi(by - 24 + ly, HH - 1);
            const int xx = clampi(bx - 24 + lx, WW - 1);
            sM[e] = sc[yy * WW + xx];
        }
        if (c + 1 < CMSK)
            __builtin_prefetch(src + (size_t)(c + 1) * HW + pix, 0, 1);
        __syncthreads();
        float acc = 0.f;
#pragma unroll
        for (int k = 0; k < KN; ++k)
            acc += wk[k] * sM[(ty + 24 + k_dy[k]) * 64 + (tx + 24 + k_dx[k])];
        dst[(size_t)c * HW + pix] = acc;
    }
}

// ---------------------------------------------------------------------------
extern "C" void kernel_launch(void* const* d_in, const int* in_sizes, int n_in,
                              void* d_out, int out_size, void* d_ws, size_t ws_size,
                              hipStream_t stream) {
    const float* imgs  = (const float*)d_in[0];   // [1,3,224,224]
    const float* masks = (const float*)d_in[1];   // [1,21,56,56]
    float* out = (float*)d_out;                   // [1,21,224,224]

    float* w = (float*)d_ws;
    size_t o = 0;
    float* bufA = w + o; o += (size_t)CMSK * HW;       // upsampled masks (ping)
    float* afft = w + o; o += (size_t)KN * HW;         // aff_total [48,224,224]
    float* aff2 = w + o; o += (size_t)KN * HW2;        // low-res raw aff
    float* img2 = w + o; o += (size_t)CIMG * HW2;      // imgs @ 67x67
    float* tmp  = w + o; o += (size_t)CMSK * HH * 56;  // GEMM row-pass temp
    float* R_a  = w + o; o += 224 * 56;                // 56 -> 224 resize matrix
    float* RT_a = w + o; o += 56 * 224;
    float* R_b  = w + o; o += 67 * 224;                // 224 -> 67 resize matrix
    float* RT_b = w + o; o += 224 * 67;
    float* posb = w + o; o += 64;                      // pos softmax (48 + pad)
    float* bufB = w + o; o += (size_t)CMSK * HW;       // pong

    // Resize matrices + positional softmax.
    const long nz = 224 * 56 + 56 * 224 + 67 * 224 + 224 * 67 + 64;
    zero_kernel<<<(nz + 255) / 256, 256, 0, stream>>>(R_a, nz);
    fill_resize<<<1, 256, 0, stream>>>(R_a, RT_a, 224, 56);
    fill_resize<<<1, 256, 0, stream>>>(R_b, RT_b, 67, 224);
    pos_kernel<<<1, 32, 0, stream>>>(posb);

    // masks 56 -> 224 (align_corners) as two WMMA GEMM passes.
    gemm_wmma_f32<<<dim3(14, 4, CMSK), 32, 0, stream>>>(
        R_a, 0, masks, 56 * 56, tmp, 224 * 56, 224, 56, 56, 56, 56, 56);
    gemm_wmma_f32<<<dim3(14, 14, CMSK), 32, 0, stream>>>(
        tmp, 224 * 56, RT_a, 0, bufA, HW, 224, 224, 56, 56, 224, 224);

    // imgs 224 -> 67 (align_corners) as two WMMA GEMM passes.
    gemm_wmma_f32<<<dim3(5, 14, CIMG), 32, 0, stream>>>(
        R_b, 0, imgs, HW, tmp, 67 * 224, 67, 224, 224, 224, 224, 224);
    gemm_wmma_f32<<<dim3(5, 5, CIMG), 32, 0, stream>>>(
        tmp, 67 * 224, RT_b, 0, img2, HW2, 67, 67, 224, 224, 67, 67);

    // Affinities and combination.
    aff_full_kernel<<<HW / 256, 256, 0, stream>>>(imgs, afft);
    aff_low_kernel<<<(HW2 + 255) / 256, 256, 0, stream>>>(img2, aff2);
    combine_kernel<<<HW / 256, 256, 0, stream>>>(aff2, posb, afft);

    // 10 propagation iterations, ping-pong, final into d_out.
    float* pp[2] = {bufA, bufB};
    for (int it = 0; it < NITER; ++it) {
        const float* s = pp[it & 1];
        float* d = (it == NITER - 1) ? out : pp[(it + 1) & 1];
        prop_kernel<<<dim3(14, 14), 256, 0, stream>>>(s, afft, d);
    }
}